// MultiModeMultiWavelengthModel_38208029065725
// MI455X (gfx1250) — compile-verified
//
#include <hip/hip_runtime.h>
#include <hip/hip_bf16.h>
#include <math.h>

// ---------------------------------------------------------------------------
// Multi-layer diffractive optical model on MI455X (gfx1250).
// 512-point DFT as dense matmul on the f32 WMMA pipe:
//   fft2(X)  = F X F,  F[k][n] = exp(-2*pi*i*k*n/512)
//   ifft2(Z) = G Z G,  G = conj(F)/512   (1/N^2 folds in, G applied twice)
// Complex GEMM with 2x2 register blocking: each wave owns a 32x32 complex
// output macro-tile, accumulating Cr/Ci directly (negated-Ai trick) with
// V_WMMA_F32_16X16X4_F32. 16 wmma : 12 vmem per k-step.
// ---------------------------------------------------------------------------

typedef float v2f __attribute__((ext_vector_type(2)));
typedef float v8f __attribute__((ext_vector_type(8)));

#define NSIZE 512
#define NBATCH 48                      // 16 modes * 3 wavelengths
#define PLANE (NSIZE * NSIZE)          // 262144 elements
#define FIELD_PLANE ((size_t)NBATCH * PLANE)

// ---------------------------------------------------------------------------
// Batched complex GEMM: C[b] = A[b] * B[b], all 512x512, separate f32 planes.
// Batch stride 0 on A or B makes that operand shared (the DFT matrix).
//
// WMMA f32 16x16x4 per-lane layouts (ISA 7.12.2):
//   A 16x4 : lane L: M = L%16; VGPR v holds K = v + 2*(L/16)
//   B 4x16 : lane L: N = L%16; VGPR v holds K = v + 2*(L/16)
//   C 16x16: lane L: N = L%16; VGPR j holds M = j + 8*(L/16)
//
// Complex product needs Ar*Br - Ai*Bi and Ar*Bi + Ai*Br; f32 WMMA NEG bits
// only allow C-negate, so the single negative product uses a VALU-negated
// copy of Ai.
// ---------------------------------------------------------------------------
__global__ __launch_bounds__(128) void cgemm_wmma_f32(
    const float* __restrict__ Ar, const float* __restrict__ Ai, long long sA,
    const float* __restrict__ Br, const float* __restrict__ Bi, long long sB,
    float* __restrict__ Cr, float* __restrict__ Ci, long long sC)
{
    const int N = NSIZE;
    const int lane = threadIdx.x & 31;
    const int wave = threadIdx.x >> 5;
    const int t    = blockIdx.x * 4 + wave;        // macro-tile id, 0..255
    const int tm   = (t >> 4) << 5;                // macroM * 32
    const int tn   = (t & 15) << 5;                // macroN * 32
    const long long b = blockIdx.y;

    Ar += b * sA;  Ai += b * sA;
    Br += b * sB;  Bi += b * sB;
    Cr += b * sC;  Ci += b * sC;

    const int half = lane >> 4;                    // 0 or 1
    const int lm   = lane & 15;

    // A pointers: rows tm..tm+31, K pair at half*2 (second M-tile via +16*N)
    const float* arp = Ar + (size_t)(tm + lm) * N + half * 2;
    const float* aip = Ai + (size_t)(tm + lm) * N + half * 2;
    // B pointers: K rows at half*2, cols tn..tn+31 (second N-tile via +16)
    const float* brp = Br + (size_t)(half * 2) * N + (tn + lm);
    const float* bip = Bi + (size_t)(half * 2) * N + (tn + lm);

    // 8 accumulator chains: Cr/Ci for each of the 2x2 tiles (64 VGPRs)
    v8f cr00 = {}, cr01 = {}, cr10 = {}, cr11 = {};
    v8f ci00 = {}, ci01 = {}, ci10 = {}, ci11 = {};

    #pragma unroll 2
    for (int k0 = 0; k0 < N; k0 += 4) {
        // --- loads: 4x b64 (A) + 8x b32 (B) for 16 wmma ---
        v2f ar0 = *(const v2f*)arp;                        // M-tile 0
        v2f ai0 = *(const v2f*)aip;
        v2f ar1 = *(const v2f*)(arp + 16 * N);             // M-tile 1
        v2f ai1 = *(const v2f*)(aip + 16 * N);

        v2f br0; br0.x = brp[0];      br0.y = brp[N];      // N-tile 0
        v2f bi0; bi0.x = bip[0];      bi0.y = bip[N];
        v2f br1; br1.x = brp[16];     br1.y = brp[N + 16]; // N-tile 1
        v2f bi1; bi1.x = bip[16];     bi1.y = bip[N + 16];

        v2f ain0 = -ai0;                                   // for -Ai*Bi
        v2f ain1 = -ai1;

        // --- tile (0,0) ---
        cr00 = __builtin_amdgcn_wmma_f32_16x16x4_f32(false, ar0,  false, br0, (short)0, cr00, false, false);
        cr00 = __builtin_amdgcn_wmma_f32_16x16x4_f32(false, ain0, false, bi0, (short)0, cr00, false, false);
        ci00 = __builtin_amdgcn_wmma_f32_16x16x4_f32(false, ar0,  false, bi0, (short)0, ci00, false, false);
        ci00 = __builtin_amdgcn_wmma_f32_16x16x4_f32(false, ai0,  false, br0, (short)0, ci00, false, false);
        // --- tile (0,1) ---
        cr01 = __builtin_amdgcn_wmma_f32_16x16x4_f32(false, ar0,  false, br1, (short)0, cr01, false, false);
        cr01 = __builtin_amdgcn_wmma_f32_16x16x4_f32(false, ain0, false, bi1, (short)0, cr01, false, false);
        ci01 = __builtin_amdgcn_wmma_f32_16x16x4_f32(false, ar0,  false, bi1, (short)0, ci01, false, false);
        ci01 = __builtin_amdgcn_wmma_f32_16x16x4_f32(false, ai0,  false, br1, (short)0, ci01, false, false);
        // --- tile (1,0) ---
        cr10 = __builtin_amdgcn_wmma_f32_16x16x4_f32(false, ar1,  false, br0, (short)0, cr10, false, false);
        cr10 = __builtin_amdgcn_wmma_f32_16x16x4_f32(false, ain1, false, bi0, (short)0, cr10, false, false);
        ci10 = __builtin_amdgcn_wmma_f32_16x16x4_f32(false, ar1,  false, bi0, (short)0, ci10, false, false);
        ci10 = __builtin_amdgcn_wmma_f32_16x16x4_f32(false, ai1,  false, br0, (short)0, ci10, false, false);
        // --- tile (1,1) ---
        cr11 = __builtin_amdgcn_wmma_f32_16x16x4_f32(false, ar1,  false, br1, (short)0, cr11, false, false);
        cr11 = __builtin_amdgcn_wmma_f32_16x16x4_f32(false, ain1, false, bi1, (short)0, cr11, false, false);
        ci11 = __builtin_amdgcn_wmma_f32_16x16x4_f32(false, ar1,  false, bi1, (short)0, ci11, false, false);
        ci11 = __builtin_amdgcn_wmma_f32_16x16x4_f32(false, ai1,  false, br1, (short)0, ci11, false, false);

        arp += 4;      aip += 4;
        brp += 4 * N;  bip += 4 * N;
    }

    // Epilogue: pure stores (Cr/Ci already combined)
    float* crp = Cr + (size_t)(tm + half * 8) * N + tn + lm;
    float* cip = Ci + (size_t)(tm + half * 8) * N + tn + lm;
    #pragma unroll
    for (int j = 0; j < 8; ++j) {
        crp[(size_t)j * N]            = cr00[j];
        crp[(size_t)j * N + 16]       = cr01[j];
        crp[(size_t)(j + 16) * N]     = cr10[j];
        crp[(size_t)(j + 16) * N + 16]= cr11[j];
        cip[(size_t)j * N]            = ci00[j];
        cip[(size_t)j * N + 16]       = ci01[j];
        cip[(size_t)(j + 16) * N]     = ci10[j];
        cip[(size_t)(j + 16) * N + 16]= ci11[j];
    }
}

// ---------------------------------------------------------------------------
// DFT matrix build: exact twiddles via (k*n) mod 512.
//   F = exp(-i*2*pi*m/512); G = conj(F)/512
// ---------------------------------------------------------------------------
__global__ void build_dft(float* __restrict__ Fr, float* __restrict__ Fi,
                          float* __restrict__ Gr, float* __restrict__ Gi)
{
    int idx = blockIdx.x * blockDim.x + threadIdx.x;
    if (idx >= PLANE) return;
    int k = idx >> 9, n = idx & 511;
    int m = (k * n) & 511;                         // exact
    float ang = (float)m * (-6.283185307179586f / 512.0f);
    float s, c;
    sincosf(ang, &s, &c);
    Fr[idx] = c;
    Fi[idx] = s;
    Gr[idx] = c * (1.0f / 512.0f);                 // conj: cos same
    Gi[idx] = -s * (1.0f / 512.0f);                // conj: -sin
}

// ---------------------------------------------------------------------------
// Band-limited angular-spectrum transfer function, per wavelength.
//   arg = (1/lam)^2 - (fx^2+fy^2); H = exp(i*2*pi*sqrt(arg)*z) if arg>0 else 0
// ---------------------------------------------------------------------------
__global__ void build_H(const float* __restrict__ lams, float z,
                        float* __restrict__ Hr, float* __restrict__ Hi)
{
    int idx = blockIdx.x * blockDim.x + threadIdx.x;
    if (idx >= 3 * PLANE) return;
    int c   = idx / PLANE;
    int rem = idx - c * PLANE;
    int h = rem >> 9, w = rem & 511;
    const float df = 1.0f / (512.0f * 8e-6f);      // fftfreq step, 1/(n*dx)
    float fy = (float)(h < 256 ? h : h - 512) * df;
    float fx = (float)(w < 256 ? w : w - 512) * df;
    float inv_lam = 1.0f / lams[c];
    float arg = inv_lam * inv_lam - (fx * fx + fy * fy);
    if (arg > 0.0f) {
        float kz  = 6.2831853071795864f * sqrtf(arg);
        float ang = kz * z;                         // float32, as in reference
        float s, co;
        sincosf(ang, &s, &co);
        Hr[idx] = co;
        Hi[idx] = s;
    } else {
        Hr[idx] = 0.0f;
        Hi[idx] = 0.0f;
    }
}

// ---------------------------------------------------------------------------
// field = x + 0i
// ---------------------------------------------------------------------------
__global__ void init_field(const float* __restrict__ x,
                           float* __restrict__ re, float* __restrict__ im,
                           long long n)
{
    long long i = (long long)blockIdx.x * blockDim.x + threadIdx.x;
    if (i >= n) return;
    re[i] = x[i];
    im[i] = 0.0f;
}

// ---------------------------------------------------------------------------
// field *= exp(i * phase[h][w] * coeff[c]) ; batch index b -> wavelength b%3
// ---------------------------------------------------------------------------
__global__ void phase_mul(float* __restrict__ re, float* __restrict__ im,
                          const float* __restrict__ phase_l,
                          const float* __restrict__ coeff_l)
{
    long long i = (long long)blockIdx.x * blockDim.x + threadIdx.x;
    if (i >= (long long)FIELD_PLANE) return;
    int b  = (int)(i / PLANE);
    int hw = (int)(i - (long long)b * PLANE);
    int c  = b % 3;
    float eff = phase_l[hw] * coeff_l[c];
    float s, co;
    sincosf(eff, &s, &co);
    float r = re[i], q = im[i];
    re[i] = r * co - q * s;
    im[i] = r * s + q * co;
}

// ---------------------------------------------------------------------------
// field *= H[c]  (H is [3,512,512] complex planes)
// ---------------------------------------------------------------------------
__global__ void h_mul(float* __restrict__ re, float* __restrict__ im,
                      const float* __restrict__ Hr, const float* __restrict__ Hi)
{
    long long i = (long long)blockIdx.x * blockDim.x + threadIdx.x;
    if (i >= (long long)FIELD_PLANE) return;
    int b  = (int)(i / PLANE);
    int hw = (int)(i - (long long)b * PLANE);
    int c  = b % 3;
    size_t hidx = (size_t)c * PLANE + hw;
    float hr = Hr[hidx], hi = Hi[hidx];
    float r = re[i], q = im[i];
    re[i] = r * hr - q * hi;
    im[i] = r * hi + q * hr;
}

// ---------------------------------------------------------------------------
// out = |field|^2
// ---------------------------------------------------------------------------
__global__ void magnitude2(const float* __restrict__ re,
                           const float* __restrict__ im,
                           float* __restrict__ out, long long n)
{
    long long i = (long long)blockIdx.x * blockDim.x + threadIdx.x;
    if (i >= n) return;
    float r = re[i], q = im[i];
    out[i] = r * r + q * q;
}

// ---------------------------------------------------------------------------
// Launch orchestration
// ---------------------------------------------------------------------------
extern "C" void kernel_launch(void* const* d_in, const int* in_sizes, int n_in,
                              void* d_out, int out_size, void* d_ws, size_t ws_size,
                              hipStream_t stream)
{
    const float* x      = (const float*)d_in[0];   // [16,3,512,512]
    const float* phases = (const float*)d_in[1];   // [4,512,512]
    const float* coeffs = (const float*)d_in[2];   // [4,3]
    const float* lams   = (const float*)d_in[3];   // [3]

    float* out = (float*)d_out;

    // Workspace layout (floats)
    float* ws = (float*)d_ws;
    const size_t FP = FIELD_PLANE;                 // 48*512*512
    float* Are = ws;                               // field ping
    float* Aim = ws + FP;
    float* Bre = ws + 2 * FP;                      // field pong
    float* Bim = ws + 3 * FP;
    float* Fr  = ws + 4 * FP;                      // forward DFT
    float* Fi  = Fr + PLANE;
    float* Gr  = Fi + PLANE;                       // conj(F)/512
    float* Gi  = Gr + PLANE;
    float* Hlr = Gi + PLANE;                       // H inter-layer [3,512,512]
    float* Hli = Hlr + 3 * (size_t)PLANE;
    float* Hfr = Hli + 3 * (size_t)PLANE;          // H final
    float* Hfi = Hfr + 3 * (size_t)PLANE;

    const long long nField = (long long)FP;
    const int PTB = 256;                                   // pointwise block
    const int gField = (int)((nField + PTB - 1) / PTB);
    const int gPlane = (PLANE + PTB - 1) / PTB;
    const int gHm    = (3 * PLANE + PTB - 1) / PTB;

    // 1) Precompute DFT matrices and transfer functions (cheap, deterministic)
    build_dft<<<gPlane, PTB, 0, stream>>>(Fr, Fi, Gr, Gi);
    build_H<<<gHm, PTB, 0, stream>>>(lams, 0.03f, Hlr, Hli);   // Z_LAYERS
    build_H<<<gHm, PTB, 0, stream>>>(lams, 0.05f, Hfr, Hfi);   // Z_PROP

    // 2) field = x + 0i
    init_field<<<gField, PTB, 0, stream>>>(x, Are, Aim, nField);

    // GEMM grid: 256 macro-tiles (32x32) per image, 4 waves/block -> 64 blocks
    dim3 gG(64, NBATCH), bG(128);
    const long long S = PLANE;   // per-image plane stride
    auto fft2 = [&](float* re, float* im, float* tre, float* tim) {
        // T = X * F  (rows)
        cgemm_wmma_f32<<<gG, bG, 0, stream>>>(re, im, S, Fr, Fi, 0, tre, tim, S);
        // X = F * T  (cols)
        cgemm_wmma_f32<<<gG, bG, 0, stream>>>(Fr, Fi, 0, tre, tim, S, re, im, S);
    };
    auto ifft2 = [&](float* re, float* im, float* tre, float* tim) {
        cgemm_wmma_f32<<<gG, bG, 0, stream>>>(re, im, S, Gr, Gi, 0, tre, tim, S);
        cgemm_wmma_f32<<<gG, bG, 0, stream>>>(Gr, Gi, 0, tre, tim, S, re, im, S);
    };

    // 3) Four phase layers, each followed by inter-layer propagation
    for (int l = 0; l < 4; ++l) {
        phase_mul<<<gField, PTB, 0, stream>>>(Are, Aim,
                                              phases + (size_t)l * PLANE,
                                              coeffs + 3 * l);
        fft2(Are, Aim, Bre, Bim);
        h_mul<<<gField, PTB, 0, stream>>>(Are, Aim, Hlr, Hli);
        ifft2(Are, Aim, Bre, Bim);
    }

    // 4) Final propagation to detector
    fft2(Are, Aim, Bre, Bim);
    h_mul<<<gField, PTB, 0, stream>>>(Are, Aim, Hfr, Hfi);
    ifft2(Are, Aim, Bre, Bim);

    // 5) Intensity
    magnitude2<<<gField, PTB, 0, stream>>>(Are, Aim, out, nField);
}